// Matcher_61572651155615
// MI455X (gfx1250) — compile-verified
//
#include <hip/hip_runtime.h>

typedef __attribute__((ext_vector_type(2))) float v2f;
typedef __attribute__((ext_vector_type(8))) float v8f;

#define BATCH 64
#define NPROP 4096
#define NGT   128

// ---------------------------------------------------------------------------
// Kernel 1: cost[b][g][q] = 1 - dot(gt[b][g], query[q]) via V_WMMA_F32_16X16X4_F32
// One wave computes a 16(g) x 16(q) tile: A = gt tile (16x4, K=3 padded),
// B = query tile (4x16). D layout gives coalesced stores along q.
// ---------------------------------------------------------------------------
__global__ __launch_bounds__(256)
void cost_wmma_kernel(const float* __restrict__ gtn,
                      const float* __restrict__ qn,
                      float* __restrict__ cost) {
  const int wave = (int)((blockIdx.x * blockDim.x + threadIdx.x) >> 5);
  const int lane = (int)(threadIdx.x & 31);
  // wave -> (b, g_tile, q_tile): 64 * 8 * 256 waves total
  const int qt = wave & 255;
  const int gt = (wave >> 8) & 7;
  const int b  = wave >> 11;
  const int g0 = gt << 4;
  const int q0 = qt << 4;
  const int idx = lane & 15;
  const int hi  = lane >> 4;   // 0: lanes 0-15 (K=0,1), 1: lanes 16-31 (K=2,3)

  const float* g = gtn + ((size_t)b * NGT + (size_t)(g0 + idx)) * 3;
  const float* q = qn  + (size_t)(q0 + idx) * 3;
  const float gx = g[0], gy = g[1], gz = g[2];
  const float qx = q[0], qy = q[1], qz = q[2];

  v2f a, bb;
  a.x  = hi ? gz : gx;  a.y  = hi ? 0.0f : gy;   // K3 pad = 0
  bb.x = hi ? qz : qx;  bb.y = hi ? 0.0f : qy;

  v8f c = {};
  c = __builtin_amdgcn_wmma_f32_16x16x4_f32(false, a, false, bb,
                                            (short)0, c, false, false);

  // D: VGPR v, lane l -> M(g) = v + 8*hi, N(q) = idx. Store 1 - D.
  float* out = cost + ((size_t)b * NGT + (size_t)(g0 + hi * 8)) * NPROP
                    + (size_t)q0 + (size_t)idx;
  #pragma unroll
  for (int v = 0; v < 8; ++v)
    out[(size_t)v * NPROP] = 1.0f - c[v];
}

// ---------------------------------------------------------------------------
// Kernel 2: Jonker-Volgenant shortest augmenting path, one workgroup per batch.
// 256 threads; columns (proposals) partitioned tid-strided for coalescing.
// Cost rows read from the L2-resident precomputed matrix (or recomputed from
// LDS gt normals if workspace was too small).
// ---------------------------------------------------------------------------
__global__ __launch_bounds__(256)
void hungarian_kernel(const float* __restrict__ gtn,
                      const float* __restrict__ qn,
                      const int*   __restrict__ nag,
                      const float* __restrict__ cost,   // may be nullptr
                      float* __restrict__ out) {
  const int b   = (int)blockIdx.x;
  const int tid = (int)threadIdx.x;

  __shared__ float v_arr[NPROP];
  __shared__ float minv [NPROP];
  __shared__ short way_arr[NPROP];
  __shared__ short p_arr [NPROP];     // column -> matched row, -1 = free
  __shared__ unsigned char used[NPROP];
  __shared__ float u_arr[NGT + 1];
  __shared__ float gts[NGT * 3];
  __shared__ float red_min[8];
  __shared__ int   red_idx[8];
  __shared__ float sh_delta;
  __shared__ int   sh_j1;
  __shared__ int   sh_brk;

  for (int j = tid; j < NPROP; j += 256) { v_arr[j] = 0.0f; p_arr[j] = -1; }
  for (int j = tid; j < NGT + 1; j += 256) u_arr[j] = 0.0f;
  for (int j = tid; j < NGT * 3; j += 256) gts[j] = gtn[(size_t)b * NGT * 3 + j];
  __syncthreads();

  const int k = nag[b];
  const float* crow_base = cost ? (cost + (size_t)b * NGT * NPROP) : nullptr;

  for (int i = 0; i < k; ++i) {
    for (int j = tid; j < NPROP; j += 256) { minv[j] = __builtin_inff(); used[j] = 0; }
    __syncthreads();

    int j0 = -1;                       // -1 == virtual source column
    while (true) {
      const int   i0  = (j0 < 0) ? i : (int)p_arr[j0];
      const float ui0 = u_arr[i0];
      if (tid == 0 && j0 >= 0) used[j0] = 1;
      __syncthreads();

      // relax from row i0 + local argmin over free columns
      float lmin = __builtin_inff();
      int   lidx = NPROP;
      const float gx = gts[i0 * 3], gy = gts[i0 * 3 + 1], gz = gts[i0 * 3 + 2];
      for (int j = tid; j < NPROP; j += 256) {
        if (!used[j]) {
          float cj;
          if (crow_base) {
            cj = crow_base[(size_t)i0 * NPROP + j];
          } else {
            const float* qv = qn + (size_t)j * 3;
            cj = 1.0f - (gx * qv[0] + gy * qv[1] + gz * qv[2]);
          }
          const float cur = cj - ui0 - v_arr[j];
          float mj = minv[j];
          if (cur < mj) { mj = cur; minv[j] = cur; way_arr[j] = (short)j0; }
          if (mj < lmin) { lmin = mj; lidx = j; }
        }
      }
      // wave32 min+argmin reduction (ties -> lowest column, matching np.argmin)
      #pragma unroll
      for (int off = 16; off > 0; off >>= 1) {
        const float om = __shfl_xor(lmin, off, 32);
        const int   oi = __shfl_xor(lidx, off, 32);
        if (om < lmin || (om == lmin && oi < lidx)) { lmin = om; lidx = oi; }
      }
      if ((tid & 31) == 0) { red_min[tid >> 5] = lmin; red_idx[tid >> 5] = lidx; }
      __syncthreads();
      if (tid == 0) {
        float dm = red_min[0]; int dj = red_idx[0];
        #pragma unroll
        for (int w = 1; w < 8; ++w)
          if (red_min[w] < dm || (red_min[w] == dm && red_idx[w] < dj)) {
            dm = red_min[w]; dj = red_idx[w];
          }
        sh_delta = dm; sh_j1 = dj;
        u_arr[i] += dm;                       // virtual column carries row i
        sh_brk = (p_arr[dj] < 0) ? 1 : 0;
      }
      __syncthreads();

      const float delta = sh_delta;
      const int   j1    = sh_j1;
      for (int j = tid; j < NPROP; j += 256) {
        if (used[j]) {
          v_arr[j] -= delta;
          u_arr[(int)p_arr[j]] += delta;      // distinct rows: race-free
        } else {
          minv[j] -= delta;
        }
      }
      const int brk = sh_brk;
      j0 = j1;
      __syncthreads();
      if (brk) break;
    }

    // augment along predecessor chain (short; serial on thread 0)
    if (tid == 0) {
      int jj = j0;
      while (true) {
        const int jp = (int)way_arr[jj];
        p_arr[jj] = (jp < 0) ? (short)i : p_arr[jp];
        if (jp < 0) break;
        jj = jp;
      }
    }
    __syncthreads();
  }

  // outputs: [0, B*NPROP) = per_prop_gt_inds (as float), then matched mask
  for (int j = tid; j < NPROP; j += 256) {
    const int r = (int)p_arr[j];
    out[(size_t)b * NPROP + j] = (r >= 0) ? (float)r : 0.0f;
    out[(size_t)BATCH * NPROP + (size_t)b * NPROP + j] = (r >= 0) ? 1.0f : 0.0f;
  }
}

// ---------------------------------------------------------------------------
extern "C" void kernel_launch(void* const* d_in, const int* in_sizes, int n_in,
                              void* d_out, int out_size, void* d_ws, size_t ws_size,
                              hipStream_t stream) {
  (void)in_sizes; (void)n_in; (void)out_size;
  // input order: cls_prob, gt_box_present, num_actual_gt, gt_normal_normalized, query_normals
  const int*   nag = (const int*)  d_in[2];
  const float* gtn = (const float*)d_in[3];
  const float* qn  = (const float*)d_in[4];
  float* out = (float*)d_out;

  const size_t need = (size_t)BATCH * NGT * NPROP * sizeof(float);  // 128 MB, fits in 192 MB L2
  float* cost = (ws_size >= need) ? (float*)d_ws : nullptr;

  if (cost) {
    // 64 batches * 8 g-tiles * 256 q-tiles = 131072 waves; 8 waves/block
    cost_wmma_kernel<<<16384, 256, 0, stream>>>(gtn, qn, cost);
  }
  hungarian_kernel<<<BATCH, 256, 0, stream>>>(gtn, qn, nag, cost, out);
}